// GravNetBlock_35974646071517
// MI455X (gfx1250) — compile-verified
//
#include <hip/hip_runtime.h>
#include <math.h>

#define NNODES 16384
#define CCH    64
#define SDIM   3
#define KNN    40

typedef float v2f __attribute__((ext_vector_type(2)));
typedef float v8f __attribute__((ext_vector_type(8)));

// ---------------------------------------------------------------------------
// Kernel 1: learned coordinates s = x @ w_s  (+ squared norms)
// ---------------------------------------------------------------------------
__global__ void __launch_bounds__(256) coords_kernel(
    const float* __restrict__ x, const float* __restrict__ w_s,
    float* __restrict__ s, float* __restrict__ sq, int n) {
  int i = blockIdx.x * blockDim.x + threadIdx.x;
  if (i >= n) return;
  float a0 = 0.f, a1 = 0.f, a2 = 0.f;
  const float* xr = x + i * CCH;
#pragma unroll 8
  for (int k = 0; k < CCH; ++k) {
    float xv = xr[k];
    a0 = fmaf(xv, w_s[k * SDIM + 0], a0);
    a1 = fmaf(xv, w_s[k * SDIM + 1], a1);
    a2 = fmaf(xv, w_s[k * SDIM + 2], a2);
  }
  s[i * SDIM + 0] = a0;
  s[i * SDIM + 1] = a1;
  s[i * SDIM + 2] = a2;
  sq[i] = a0 * a0 + a1 * a1 + a2 * a2;
}

// ---------------------------------------------------------------------------
// Kernel 2 (reused 4x): fp32 WMMA GEMM  Out[M x 64] = A[M x KD] @ W[KD x 64]
// epilogue: optional bias[64], ELU, residual R[M x 64]
// One wave -> one 16x16 tile; 8 waves per 256-thread block.
// ---------------------------------------------------------------------------
template <int KD, bool BIAS, bool ELU, bool RES>
__global__ void __launch_bounds__(256) gemm_wmma_kernel(
    const float* __restrict__ A, const float* __restrict__ W,
    const float* __restrict__ bias, const float* __restrict__ R,
    float* __restrict__ Out, int M) {
  const int lane  = threadIdx.x & 31;
  const int wave  = threadIdx.x >> 5;
  const int tile  = blockIdx.x * 8 + wave;      // total tiles = (M/16)*4
  const int tm    = tile >> 2;                  // row tile
  const int tn    = tile & 3;                   // col tile (64 cols -> 4 tiles)
  const int l15   = lane & 15;
  const int khalf = (lane >> 4) * 2;            // lanes 16-31 carry K+2,K+3
  const int row   = tm * 16 + l15;              // A row for this lane
  const int col   = tn * 16 + l15;              // B/D column for this lane

  v8f acc = {};
  const float* Ar = A + row * KD;
#pragma unroll 4
  for (int k = 0; k < KD; k += 4) {
    v2f a, b;
    a.x = Ar[k + khalf];
    a.y = Ar[k + khalf + 1];
    b.x = W[(k + khalf) * CCH + col];
    b.y = W[(k + khalf + 1) * CCH + col];
    acc = __builtin_amdgcn_wmma_f32_16x16x4_f32(
        false, a, false, b, (short)0, acc, false, false);
  }

  const int rbase = tm * 16 + (lane >> 4) * 8;  // D rows: lanes>=16 hold M+8
  float bv = BIAS ? bias[col] : 0.0f;
#pragma unroll
  for (int r = 0; r < 8; ++r) {
    int orow = rbase + r;
    float v = acc[r];
    if (BIAS) v += bv;
    if (ELU)  v = v > 0.0f ? v : expm1f(v);
    if (RES)  v += R[orow * CCH + col];
    Out[orow * CCH + col] = v;
  }
}

// ---------------------------------------------------------------------------
// Kernel 3: exact brute-force kNN (K=40) per node + edge weights.
// One 512-thread block per node; 32 (d2,idx) keys per thread in registers;
// 40 rounds of "min key strictly greater than last" (lexicographic).
// ---------------------------------------------------------------------------
__device__ __forceinline__ unsigned long long shfl_xor_u64(unsigned long long v,
                                                           int mask) {
  unsigned lo = (unsigned)v, hi = (unsigned)(v >> 32);
  lo = __shfl_xor((unsigned)lo, mask, 32);
  hi = __shfl_xor((unsigned)hi, mask, 32);
  return ((unsigned long long)hi << 32) | lo;
}

__global__ void __launch_bounds__(512) knn_kernel(
    const float* __restrict__ s, const float* __restrict__ sq,
    int* __restrict__ idx_out, float* __restrict__ ew_out, int n) {
  const int i = blockIdx.x;
  const int t = threadIdx.x;
  const float si0 = s[i * SDIM + 0];
  const float si1 = s[i * SDIM + 1];
  const float si2 = s[i * SDIM + 2];
  const float sqi = sq[i];

  unsigned long long key[32];
#pragma unroll
  for (int c = 0; c < 32; ++c) {
    int j = t + 512 * c;
    float dot = si0 * s[j * SDIM + 0] + si1 * s[j * SDIM + 1] +
                si2 * s[j * SDIM + 2];
    float d2 = fmaxf(sqi + sq[j] - 2.0f * dot, 0.0f);
    // non-negative float bit pattern is order-preserving; +1 bias on index
    // makes every key > 0 and keeps ascending-index tie-break.
    key[c] = ((unsigned long long)__float_as_uint(d2) << 32) |
             (unsigned int)(j + 1);
  }

  __shared__ unsigned long long red[16];
  __shared__ unsigned long long bcast;
  unsigned long long last = 0ull;

  for (int k = 0; k < KNN; ++k) {
    unsigned long long best = ~0ull;
#pragma unroll
    for (int c = 0; c < 32; ++c)
      if (key[c] > last && key[c] < best) best = key[c];
    // wave32 min-reduction
    for (int off = 16; off > 0; off >>= 1) {
      unsigned long long o = shfl_xor_u64(best, off);
      if (o < best) best = o;
    }
    if ((t & 31) == 0) red[t >> 5] = best;
    __syncthreads();
    if (t == 0) {
      unsigned long long b = red[0];
#pragma unroll
      for (int w = 1; w < 16; ++w)
        if (red[w] < b) b = red[w];
      bcast = b;
      unsigned int j = (unsigned int)(b & 0xffffffffu) - 1u;
      float d0 = si0 - s[j * SDIM + 0];
      float d1 = si1 - s[j * SDIM + 1];
      float d2v = si2 - s[j * SDIM + 2];
      float dd = d0 * d0 + d1 * d1 + d2v * d2v;   // exact, as reference
      idx_out[i * KNN + k] = (int)j;
      ew_out[i * KNN + k] = expf(-(dd + 1e-6f));
    }
    __syncthreads();
    last = bcast;
  }
}

// ---------------------------------------------------------------------------
// Kernel 4: mailbox aggregation -> xcat = [mean(msg), max(msg), x]  (N x 192)
// ---------------------------------------------------------------------------
__global__ void __launch_bounds__(256) aggregate_kernel(
    const float* __restrict__ h, const float* __restrict__ x,
    const int* __restrict__ idx, const float* __restrict__ ew,
    float* __restrict__ xcat, int n) {
  int gid = blockIdx.x * blockDim.x + threadIdx.x;
  int i = gid >> 6;
  int c = gid & 63;
  if (i >= n) return;
  float sum = 0.0f, mx = -INFINITY;
#pragma unroll 4
  for (int k = 0; k < KNN; ++k) {
    int j = idx[i * KNN + k];
    float w = ew[i * KNN + k];
    float v = w * h[j * CCH + c];
    sum += v;
    mx = fmaxf(mx, v);
  }
  xcat[i * 3 * CCH + c]            = sum * (1.0f / (float)KNN);
  xcat[i * 3 * CCH + CCH + c]      = mx;
  xcat[i * 3 * CCH + 2 * CCH + c]  = x[i * CCH + c];
}

// ---------------------------------------------------------------------------
// Kernel 5: per-channel biased batch stats -> mu[c], rsqrt(var+eps)[c]
// ---------------------------------------------------------------------------
__global__ void __launch_bounds__(256) bn_stats_kernel(
    const float* __restrict__ t, float* __restrict__ mu,
    float* __restrict__ rinv, int n) {
  __shared__ float ssum[256];
  __shared__ float ssq[256];
  int c = blockIdx.x;
  float s = 0.0f, s2 = 0.0f;
  for (int r = threadIdx.x; r < n; r += 256) {
    float v = t[r * CCH + c];
    s += v;
    s2 += v * v;
  }
  ssum[threadIdx.x] = s;
  ssq[threadIdx.x] = s2;
  __syncthreads();
  for (int off = 128; off > 0; off >>= 1) {
    if (threadIdx.x < off) {
      ssum[threadIdx.x] += ssum[threadIdx.x + off];
      ssq[threadIdx.x]  += ssq[threadIdx.x + off];
    }
    __syncthreads();
  }
  if (threadIdx.x == 0) {
    float m = ssum[0] / (float)n;
    float var = ssq[0] / (float)n - m * m;
    mu[c] = m;
    rinv[c] = rsqrtf(var + 1e-5f);
  }
}

// ---------------------------------------------------------------------------
// Kernel 6: apply batchnorm (training-mode) elementwise
// ---------------------------------------------------------------------------
__global__ void __launch_bounds__(256) bn_apply_kernel(
    const float* __restrict__ t, const float* __restrict__ mu,
    const float* __restrict__ rinv, const float* __restrict__ gamma,
    const float* __restrict__ beta, float* __restrict__ out, int total) {
  int i = blockIdx.x * blockDim.x + threadIdx.x;
  if (i >= total) return;
  int c = i & 63;
  out[i] = gamma[c] * (t[i] - mu[c]) * rinv[c] + beta[c];
}

// ---------------------------------------------------------------------------
// Host orchestration
// ---------------------------------------------------------------------------
extern "C" void kernel_launch(void* const* d_in, const int* in_sizes, int n_in,
                              void* d_out, int out_size, void* d_ws,
                              size_t ws_size, hipStream_t stream) {
  (void)in_sizes; (void)n_in; (void)out_size; (void)ws_size;
  const int N = NNODES;

  const float* x      = (const float*)d_in[0];
  const float* w_s    = (const float*)d_in[1];
  const float* w_h    = (const float*)d_in[2];
  const float* b_h    = (const float*)d_in[3];
  const float* w_lin  = (const float*)d_in[4];
  const float* b_lin  = (const float*)d_in[5];
  const float* w_p1   = (const float*)d_in[6];
  const float* b_p1   = (const float*)d_in[7];
  const float* w_p2   = (const float*)d_in[8];
  const float* b_p2   = (const float*)d_in[9];
  const float* gamma2 = (const float*)d_in[10];
  const float* beta2  = (const float*)d_in[11];
  const float* gamma3 = (const float*)d_in[12];
  const float* beta3  = (const float*)d_in[13];
  float* out = (float*)d_out;

  // workspace carve-up (256B aligned)
  char* p = (char*)d_ws;
  auto carve = [&](size_t bytes) -> void* {
    void* q = (void*)p;
    p += (bytes + 255) & ~(size_t)255;
    return q;
  };
  float* s    = (float*)carve((size_t)N * SDIM * 4);
  float* sq   = (float*)carve((size_t)N * 4);
  float* h    = (float*)carve((size_t)N * CCH * 4);
  int*   idx  = (int*)  carve((size_t)N * KNN * 4);
  float* ew   = (float*)carve((size_t)N * KNN * 4);
  float* xcat = (float*)carve((size_t)N * 3 * CCH * 4);
  float* t1   = (float*)carve((size_t)N * CCH * 4);
  float* y    = (float*)carve((size_t)N * CCH * 4);
  float* z1   = (float*)carve((size_t)N * CCH * 4);
  float* t2   = (float*)carve((size_t)N * CCH * 4);
  float* mu2  = (float*)carve(CCH * 4);
  float* ri2  = (float*)carve(CCH * 4);
  float* mu3  = (float*)carve(CCH * 4);
  float* ri3  = (float*)carve(CCH * 4);

  const int gemm_blocks = N / 32;  // (N/16 row tiles * 4 col tiles) / 8 waves

  // s = x @ w_s ; sq = |s|^2
  coords_kernel<<<N / 256, 256, 0, stream>>>(x, w_s, s, sq, N);
  // h = x @ w_h + b_h
  gemm_wmma_kernel<CCH, true, false, false>
      <<<gemm_blocks, 256, 0, stream>>>(x, w_h, b_h, nullptr, h, N);
  // kNN in learned space + edge weights
  knn_kernel<<<N, 512, 0, stream>>>(s, sq, idx, ew, N);
  // xcat = [mean(msg), max(msg), x]
  aggregate_kernel<<<(N * CCH) / 256, 256, 0, stream>>>(h, x, idx, ew, xcat, N);
  // t1 = xcat @ w_lin + b_lin + x   (pre-BN2 residual)
  gemm_wmma_kernel<3 * CCH, true, false, true>
      <<<gemm_blocks, 256, 0, stream>>>(xcat, w_lin, b_lin, x, t1, N);
  // y = BN2(t1)
  bn_stats_kernel<<<CCH, 256, 0, stream>>>(t1, mu2, ri2, N);
  bn_apply_kernel<<<(N * CCH) / 256, 256, 0, stream>>>(t1, mu2, ri2, gamma2,
                                                       beta2, y, N * CCH);
  // z1 = elu(y @ w_p1 + b_p1)
  gemm_wmma_kernel<CCH, true, true, false>
      <<<gemm_blocks, 256, 0, stream>>>(y, w_p1, b_p1, nullptr, z1, N);
  // t2 = z1 @ w_p2 + b_p2 + y   (pre-BN3 residual)
  gemm_wmma_kernel<CCH, true, false, true>
      <<<gemm_blocks, 256, 0, stream>>>(z1, w_p2, b_p2, y, t2, N);
  // out = BN3(t2)
  bn_stats_kernel<<<CCH, 256, 0, stream>>>(t2, mu3, ri3, N);
  bn_apply_kernel<<<(N * CCH) / 256, 256, 0, stream>>>(t2, mu3, ri3, gamma3,
                                                       beta3, out, N * CCH);
}